// LabelSage_57844619542595
// MI455X (gfx1250) — compile-verified
//
#include <hip/hip_runtime.h>
#include <hip/hip_bf16.h>

typedef float v2f __attribute__((ext_vector_type(2)));
typedef float v8f __attribute__((ext_vector_type(8)));

#define IN_F   128
#define OUT_F  128
#define CLS    40
#define CLS_P  48   // padded to 3 tiles of 16

#define TS_PITCH 132                       // floats per t-strip row (bank-stagger pad)
#define WBUF_BYTES (32 * 8 * 32 * 8)       // 64 KB packed B fragments (at LDS offset 0)
#define TS_BYTES   (8 * 16 * TS_PITCH * 4) // 66 KB per-wave t strips

// ---------------------------------------------------------------------------
// Kernel 1: zero the workspace (agg[N*128] + deg[N])
// ---------------------------------------------------------------------------
__global__ __launch_bounds__(256) void sage_zero(float* __restrict__ p, long n) {
    long i = (long)blockIdx.x * 256 + threadIdx.x;
    if (i < n) p[i] = 0.0f;
}

// ---------------------------------------------------------------------------
// Kernel 2: edge scatter-add. One wave (32 lanes) per edge; lane handles 4 cols.
// ---------------------------------------------------------------------------
__global__ __launch_bounds__(256) void sage_scatter(const float* __restrict__ feat,
                                                    const int* __restrict__ src,
                                                    const int* __restrict__ dst,
                                                    float* __restrict__ agg,
                                                    float* __restrict__ deg,
                                                    int E) {
    long gid = (long)blockIdx.x * 256 + threadIdx.x;
    int e    = (int)(gid >> 5);
    int lane = (int)(gid & 31);
    if (e >= E) return;
    int s = src[e];
    int d = dst[e];
    const float4 v = *(const float4*)(feat + (size_t)s * IN_F + lane * 4);
    float* ap = agg + (size_t)d * IN_F + lane * 4;
    atomicAdd(ap + 0, v.x);
    atomicAdd(ap + 1, v.y);
    atomicAdd(ap + 2, v.z);
    atomicAdd(ap + 3, v.w);
    if (lane == 0) atomicAdd(deg + d, 1.0f);
}

// ---------------------------------------------------------------------------
// Kernel 3: agg -> h_neigh = agg / max(deg,1)   (in place, float4 per thread)
// ---------------------------------------------------------------------------
__global__ __launch_bounds__(256) void sage_mean(float* __restrict__ agg,
                                                 const float* __restrict__ deg,
                                                 int N) {
    long i = (long)blockIdx.x * 256 + threadIdx.x;          // over N*32 float4 groups
    long total = (long)N * (IN_F / 4);
    if (i >= total) return;
    int node = (int)(i >> 5);
    float dg = deg[node];
    dg = dg > 1.0f ? dg : 1.0f;
    float inv = 1.0f / dg;
    float4* p = (float4*)agg + i;
    float4 v = *p;
    v.x *= inv; v.y *= inv; v.z *= inv; v.w *= inv;
    *p = v;
}

// ---------------------------------------------------------------------------
// Kernel 4: fused WMMA GEMMs with LDS-staged, frag-packed weights.
//   t   = relu(feat @ W_self + h_neigh @ W_neigh + b_neigh)   [N,128]
//   out = t @ W_pred + b_pred                                 [N,40]
// Block = 256 threads = 8 waves; each wave owns a 16-row strip; block = 128 rows.
//
// Manual LDS layout (single block) so the packed weight buffer sits at LDS
// offset 0: all wbuf fragment addresses are < 64 KB and the nt*256 component
// folds into the ds_load_b64 16-bit immediate offset (kills the per-tile
// v_add_nc_u32 address math the automatic allocator forced).
//
// V_WMMA_F32_16X16X4_F32 fragment layout (ISA 7.12.2):
//   A (16x4):  lane l<16 -> row M=l, K={k0,k0+1}; lane>=16 -> row M=l-16, K={k0+2,k0+3}
//   B (4x16):  lane l holds {W[kk][col], W[kk+1][col]}, kk=k0+(l>>4)*2, col=nt*16+(l&15)
//   C/D:       elem r of v8f -> row M = r + (lane>=16 ? 8 : 0), col = nt*16 + (lane&15)
// ---------------------------------------------------------------------------
__global__ __launch_bounds__(256) void sage_gemm(const float* __restrict__ feat,
                                                 const float* __restrict__ hne,
                                                 const float* __restrict__ Wself,
                                                 const float* __restrict__ Wneigh,
                                                 const float* __restrict__ bneigh,
                                                 const float* __restrict__ Wpred,
                                                 const float* __restrict__ bpred,
                                                 float* __restrict__ out,   // [N,40]
                                                 float* __restrict__ tout,  // [N,128]
                                                 int N) {
    __shared__ __align__(16) char lds_raw[WBUF_BYTES + TS_BYTES];
    v2f*   wbuf = (v2f*)lds_raw;                      // [k4][nt][lane] -> k4*256+nt*32+lane
    float* tsf  = (float*)(lds_raw + WBUF_BYTES);     // [wave][16][TS_PITCH]

    const int tid   = threadIdx.x;
    const int wave  = tid >> 5;
    const int lane  = tid & 31;
    const int lhalf = lane >> 4;             // 0|1 : selects K pair / row half
    const int lmod  = lane & 15;

    const int row0  = blockIdx.x * 128 + wave * 16;
    int arow  = row0 + lmod;                 // A-fragment source row for this lane
    int arowc = arow < N ? arow : (N - 1);   // clamp (stores are predicated)

    v8f acc[8] = {};

    // ---- GEMM 1: K = 256 = [feat | h_neigh], B = [W_self ; W_neigh] -------
    for (int half = 0; half < 2; ++half) {
        const float* A = (half == 0 ? feat : hne) + (size_t)arowc * IN_F;
        const float* W = (half == 0 ? Wself : Wneigh);

        // cooperative packed fill: 8192 v2f entries, 32 per thread
        if (half != 0) __syncthreads();      // waves done reading previous wbuf
        for (int j = 0; j < 32; ++j) {
            const int i   = tid + 256 * j;   // 0..8191
            const int le  = i & 31;
            const int nt  = (i >> 5) & 7;
            const int k4  = i >> 8;
            const int kk  = k4 * 4 + ((le >> 4) << 1);
            const int col = nt * 16 + (le & 15);
            v2f v;
            v.x = W[(size_t)kk * OUT_F + col];
            v.y = W[(size_t)(kk + 1) * OUT_F + col];
            wbuf[k4 * 256 + nt * 32 + le] = v;
        }
        __syncthreads();

        for (int k4 = 0; k4 < 32; ++k4) {
            v2f afrag = *(const v2f*)(A + k4 * 4 + lhalf * 2);  // one b64 / lane
            const v2f* wrow = wbuf + k4 * 256 + lane;           // base reg; nt*32 -> imm
            #pragma unroll
            for (int nt = 0; nt < 8; ++nt) {
                v2f bfrag = wrow[nt * 32];                      // ds_load_b64 w/ imm offs
                acc[nt] = __builtin_amdgcn_wmma_f32_16x16x4_f32(
                    false, afrag, false, bfrag, (short)0, acc[nt], false, false);
            }
        }
    }

    // ---- bias + ReLU, write t to d_out region and stage in LDS ------------
    float* tsw = tsf + wave * 16 * TS_PITCH;
    #pragma unroll
    for (int nt = 0; nt < 8; ++nt) {
        const int col = nt * 16 + lmod;
        const float bb = bneigh[col];
        #pragma unroll
        for (int r = 0; r < 8; ++r) {
            const int m = r + lhalf * 8;
            float v = acc[nt][r] + bb;
            v = v > 0.0f ? v : 0.0f;
            tsw[m * TS_PITCH + col] = v;
            const int row = row0 + m;
            if (row < N) tout[(size_t)row * OUT_F + col] = v;
        }
    }
    __syncthreads();                          // all waves done with wbuf (W_neigh)

    // ---- pack W_pred (40 cols, zero-padded to 48) into wbuf ---------------
    for (int j = 0; j < 12; ++j) {
        const int i   = tid + 256 * j;        // 0..3071
        const int le  = i & 31;
        const int rem = i >> 5;               // 0..95
        const int nt  = rem % 3;
        const int k4  = rem / 3;
        const int kk  = k4 * 4 + ((le >> 4) << 1);
        const int col = nt * 16 + (le & 15);
        v2f v;
        v.x = (col < CLS) ? Wpred[(size_t)kk * CLS + col]       : 0.0f;
        v.y = (col < CLS) ? Wpred[(size_t)(kk + 1) * CLS + col] : 0.0f;
        wbuf[k4 * 256 + nt * 32 + le] = v;
    }
    __syncthreads();

    // ---- GEMM 2: out = t @ W_pred + b_pred --------------------------------
    const float* tsa = tsw + lmod * TS_PITCH + lhalf * 2;   // A-frag base for this lane
    #pragma unroll
    for (int nt = 0; nt < CLS_P / 16; ++nt) {
        v8f acc2 = {};
        const int col = nt * 16 + lmod;
        const bool cok = col < CLS;
        const v2f* wrow = wbuf + nt * 32 + lane;            // k4*256 -> imm offsets
        for (int k4 = 0; k4 < 32; ++k4) {
            v2f afrag = *(const v2f*)(tsa + k4 * 4);
            v2f bfrag = wrow[k4 * 256];
            acc2 = __builtin_amdgcn_wmma_f32_16x16x4_f32(
                false, afrag, false, bfrag, (short)0, acc2, false, false);
        }
        if (cok) {
            const float bb = bpred[col];
            #pragma unroll
            for (int r = 0; r < 8; ++r) {
                const int row = row0 + r + lhalf * 8;
                if (row < N) out[(size_t)row * CLS + col] = acc2[r] + bb;
            }
        }
    }
}

// ---------------------------------------------------------------------------
extern "C" void kernel_launch(void* const* d_in, const int* in_sizes, int n_in,
                              void* d_out, int out_size, void* d_ws, size_t ws_size,
                              hipStream_t stream) {
    const float* feat   = (const float*)d_in[0];
    const int*   src    = (const int*)  d_in[1];
    const int*   dst    = (const int*)  d_in[2];
    const float* Wself  = (const float*)d_in[3];
    const float* Wneigh = (const float*)d_in[4];
    const float* bneigh = (const float*)d_in[5];
    const float* Wpred  = (const float*)d_in[6];
    const float* bpred  = (const float*)d_in[7];

    const int N = in_sizes[0] / IN_F;
    const int E = in_sizes[1];

    float* out  = (float*)d_out;                       // [N,40]
    float* tout = (float*)d_out + (size_t)N * CLS;     // [N,128]

    float* agg = (float*)d_ws;                         // [N,128]
    float* deg = agg + (size_t)N * IN_F;               // [N]

    // 1) zero agg + deg
    {
        long n = (long)N * IN_F + N;
        long blocks = (n + 255) / 256;
        sage_zero<<<(unsigned)blocks, 256, 0, stream>>>(agg, n);
    }
    // 2) edge scatter-add (wave per edge)
    {
        long threads = (long)E * 32;
        long blocks = (threads + 255) / 256;
        sage_scatter<<<(unsigned)blocks, 256, 0, stream>>>(feat, src, dst, agg, deg, E);
    }
    // 3) mean normalize in place
    {
        long n = (long)N * (IN_F / 4);
        long blocks = (n + 255) / 256;
        sage_mean<<<(unsigned)blocks, 256, 0, stream>>>(agg, deg, N);
    }
    // 4) fused WMMA GEMMs
    {
        int blocks = (N + 127) / 128;
        sage_gemm<<<blocks, 256, 0, stream>>>(feat, agg, Wself, Wneigh, bneigh,
                                              Wpred, bpred, out, tout, N);
    }
}